// PearsonEmbeddingLoss_59545426591916
// MI455X (gfx1250) — compile-verified
//
#include <hip/hip_runtime.h>

typedef __attribute__((ext_vector_type(2))) float v2f;
typedef __attribute__((ext_vector_type(8))) float v8f;

#define D_DIM 4096
#define WAVES_PER_BLOCK 8
#define MARGIN 0.0f

// ---------------------------------------------------------------------------
// Kernel 1: one wave32 per row. Streaming b128 loads, 5 running sums per lane,
// ds-swizzle butterfly reduction, per-row Pearson loss -> row_loss[row].
// ---------------------------------------------------------------------------
__global__ __launch_bounds__(256) void pearson_row_kernel(
    const float* __restrict__ x1, const float* __restrict__ x2,
    const int* __restrict__ target, float* __restrict__ row_loss, int B)
{
    const int lane = threadIdx.x & 31;
    const int wave = threadIdx.x >> 5;
    const int row  = blockIdx.x * WAVES_PER_BLOCK + wave;
    if (row >= B) return;  // whole-wave uniform exit

    const float4* p1 = (const float4*)(x1 + (size_t)row * D_DIM);
    const float4* p2 = (const float4*)(x2 + (size_t)row * D_DIM);

    float s1 = 0.f, s2 = 0.f, q1 = 0.f, q2 = 0.f, pp = 0.f;

    // 4096 cols / (32 lanes * 4 floats) = 32 iterations; unroll for MLP.
    #pragma unroll 8
    for (int it = 0; it < D_DIM / 128; ++it) {
        float4 a = p1[it * 32 + lane];   // global_load_b128, coalesced
        float4 b = p2[it * 32 + lane];
        s1 += a.x + a.y + a.z + a.w;
        s2 += b.x + b.y + b.z + b.w;
        q1 = fmaf(a.x, a.x, fmaf(a.y, a.y, fmaf(a.z, a.z, fmaf(a.w, a.w, q1))));
        q2 = fmaf(b.x, b.x, fmaf(b.y, b.y, fmaf(b.z, b.z, fmaf(b.w, b.w, q2))));
        pp = fmaf(a.x, b.x, fmaf(a.y, b.y, fmaf(a.z, b.z, fmaf(a.w, b.w, pp))));
    }

    // wave32 butterfly reduction (width 32 on gfx1250)
    #pragma unroll
    for (int off = 16; off > 0; off >>= 1) {
        s1 += __shfl_xor(s1, off, 32);
        s2 += __shfl_xor(s2, off, 32);
        q1 += __shfl_xor(q1, off, 32);
        q2 += __shfl_xor(q2, off, 32);
        pp += __shfl_xor(pp, off, 32);
    }

    if (lane == 0) {
        const float n   = (float)D_DIM;
        float num = pp - s1 * s2 / n;
        float t1  = q1 - s1 * s1 / n;
        float t2  = q2 - s2 * s2 / n;
        float den = sqrtf(t1 * t2);
        float score = (den == 0.f) ? 0.f : num / den;
        float loss  = (target[row] == 1) ? (1.f - score)
                                         : fmaxf(0.f, score - MARGIN);
        row_loss[row] = loss;
    }
}

// ---------------------------------------------------------------------------
// Kernel 2: single wave32 reduces B row losses to the mean using
// V_WMMA_F32_16X16X4_F32 with A = ones:  D[m][n] = sum_k B[k][n] (+C).
// Each WMMA folds 64 fp32 values (2 B-VGPRs x 32 lanes); total sum is the
// sum over the 16 column-sums (the B lane mapping is a permutation, which is
// irrelevant for a full commutative sum).
// ---------------------------------------------------------------------------
__global__ __launch_bounds__(32) void mean_wmma_kernel(
    const float* __restrict__ v, float* __restrict__ out, int B)
{
    const int lane = threadIdx.x;  // 0..31, EXEC all ones (WMMA requirement)

    v2f a; a[0] = 1.0f; a[1] = 1.0f;   // A = ones(16x4)
    v8f c = {};                        // C accumulator starts at zero

    const int tiles = B / 64;
    for (int t = 0; t < tiles; ++t) {
        v2f b;
        b[0] = v[t * 64 + lane];
        b[1] = v[t * 64 + 32 + lane];
        // 8 args: (neg_a, A, neg_b, B, c_mod, C, reuse_a, reuse_b)
        c = __builtin_amdgcn_wmma_f32_16x16x4_f32(
                false, a, false, b, (short)0, c, false, false);
    }

    // Tail (B not multiple of 64) via plain lane sums.
    float tail = 0.f;
    for (int i = tiles * 64 + lane; i < B; i += 32) tail += v[i];
    #pragma unroll
    for (int off = 16; off > 0; off >>= 1) tail += __shfl_xor(tail, off, 32);

    // c[0] lanes 0..15 hold colsum[n] (M=0 row); lanes 16..31 duplicate them
    // (M=8 row, identical since A rows are all ones). Reduce within 16 lanes.
    float x = c[0];
    #pragma unroll
    for (int off = 8; off > 0; off >>= 1) x += __shfl_xor(x, off, 32);

    if (lane == 0) out[0] = (x + tail) / (float)B;
}

// ---------------------------------------------------------------------------
extern "C" void kernel_launch(void* const* d_in, const int* in_sizes, int n_in,
                              void* d_out, int out_size, void* d_ws, size_t ws_size,
                              hipStream_t stream) {
    const float* x1     = (const float*)d_in[0];
    const float* x2     = (const float*)d_in[1];
    const int*   target = (const int*)d_in[2];
    float*       out    = (float*)d_out;
    float*       rowbuf = (float*)d_ws;      // B floats of scratch

    const int B = in_sizes[2];               // 4096 rows
    const int blocks = (B + WAVES_PER_BLOCK - 1) / WAVES_PER_BLOCK;

    pearson_row_kernel<<<blocks, 256, 0, stream>>>(x1, x2, target, rowbuf, B);
    mean_wmma_kernel<<<1, 32, 0, stream>>>(rowbuf, out, B);
}